// GCN_27968827031806
// MI455X (gfx1250) — compile-verified
//
#include <hip/hip_runtime.h>

#define N_NODESC 100000
#define N_EDGESC 3200000
#define F_INC    512
#define HIDC     16
#define NCLSC    2
#define BN_EPSF  1e-5f

typedef float v2f __attribute__((ext_vector_type(2)));
typedef float v8f __attribute__((ext_vector_type(8)));

// LDS aperture: generic address low 32 bits ARE the LDS byte offset (ISA §10.2).
__device__ __forceinline__ unsigned lds_off(const void* p) {
    return (unsigned)(unsigned long long)p;
}

// ---------------------------------------------------------------- utilities
__global__ void k_fill(float* __restrict__ p, float v, int n) {
    int i = blockIdx.x * blockDim.x + threadIdx.x;
    if (i < n) p[i] = v;
}

// deg[dst] += 1 per edge (deg pre-filled with 1.0 for self loops)
__global__ void k_degree(const long long* __restrict__ ei, float* __restrict__ deg) {
    int e = blockIdx.x * blockDim.x + threadIdx.x;
    if (e < N_EDGESC) {
        int d = (int)ei[(long long)N_EDGESC + e];
        atomicAdd(&deg[d], 1.0f);
    }
}

__global__ void k_rsqrt(float* __restrict__ deg) {
    int i = blockIdx.x * blockDim.x + threadIdx.x;
    if (i < N_NODESC) deg[i] = rsqrtf(deg[i]);
}

// per-edge: cast indices to i32 and compute norm = dinv[src]*dinv[dst] ONCE
__global__ void k_edgeprep(const long long* __restrict__ ei, const float* __restrict__ dinv,
                           int* __restrict__ esrc, int* __restrict__ edst,
                           float* __restrict__ enorm) {
    int e = blockIdx.x * blockDim.x + threadIdx.x;
    if (e < N_EDGESC) {
        int s = (int)ei[e];
        int d = (int)ei[(long long)N_EDGESC + e];
        esrc[e] = s;
        edst[e] = d;
        enorm[e] = dinv[s] * dinv[d];
    }
}

// ------------------------------------------------- GEMM1: hw = x @ W1  (WMMA f32)
// 8 waves/block, 128 rows/block. A tile is streamed global->LDS with
// double-buffered global_load_async_to_lds_b128 (ASYNCcnt), B (W1) resides in LDS.
__global__ __launch_bounds__(256) void k_gemm1(const float* __restrict__ x,
                                               const float* __restrict__ W1,
                                               float* __restrict__ hw) {
    constexpr int KC   = 16;                 // K per chunk
    constexpr int ROWS = 128;                // rows per block
    constexpr int APAD = 20;                 // padded row stride (floats): 80B, 16B-aligned,
                                             // bank-conflict-free for 16-lane b64 column reads
    constexpr int NCH  = F_INC / KC;         // 32 chunks

    __shared__ float sW[F_INC * HIDC];       // 32 KB
    __shared__ float sA[2][ROWS * APAD];     // 2 x 10 KB

    const int tid   = threadIdx.x;
    const int row0b = blockIdx.x * ROWS;

    // stage W1 into LDS (float4, coalesced)
    {
        const float4* w4 = (const float4*)W1;
        float4* s4 = (float4*)sW;
        for (int i = tid; i < (F_INC * HIDC) / 4; i += 256) s4[i] = w4[i];
    }

    // per-thread async src/dst: chunk = 128 rows x 16 floats = 512 float4; 2 per thread
    unsigned           lbase[2];
    unsigned long long gbase[2];
#pragma unroll
    for (int j = 0; j < 2; ++j) {
        int idx  = tid + j * 256;
        int row  = idx >> 2;                 // 4 float4 per row
        int col4 = idx & 3;
        int grow = row0b + row;
        if (grow > N_NODESC - 1) grow = N_NODESC - 1;   // clamp (stores guarded later)
        gbase[j] = (unsigned long long)(const void*)(x + (size_t)grow * F_INC)
                   + (unsigned long long)(col4 * 16);
        lbase[j] = lds_off(&sA[0][0]) + (unsigned)(row * APAD * 4 + col4 * 16);
    }
    const unsigned bufStride = (unsigned)(ROWS * APAD * 4);

    auto issue = [&](int chunk) {
        const unsigned           boff = (unsigned)(chunk & 1) * bufStride;
        const unsigned long long koff = (unsigned long long)chunk * (KC * 4);
#pragma unroll
        for (int j = 0; j < 2; ++j) {
            unsigned           ldsa = lbase[j] + boff;
            unsigned long long ga   = gbase[j] + koff;
            asm volatile("global_load_async_to_lds_b128 %0, %1, off"
                         :: "v"(ldsa), "v"(ga) : "memory");
        }
    };

    issue(0);   // prime the pipeline

    const int lane = tid & 31;
    const int wave = tid >> 5;
    const int half = lane >> 4;              // 0: K%4 in {0,1}, 1: {2,3}
    const int l16  = lane & 15;
    const int arow = wave * 16 + l16;        // this lane's A row within the block

    v8f c = {};
    for (int ch = 0; ch < NCH; ++ch) {
        asm volatile("s_wait_asynccnt 0x0" ::: "memory");  // own async writes landed
        __syncthreads();                                    // everyone's chunk visible
        if (ch + 1 < NCH) issue(ch + 1);                    // overlap next DMA w/ compute
        const float* sAb = &sA[ch & 1][0];
#pragma unroll
        for (int kk = 0; kk < KC; kk += 4) {
            v2f a = *(const v2f*)(sAb + arow * APAD + kk + 2 * half);
            const int kg = ch * KC + kk + 2 * half;
            v2f b;
            b.x = sW[kg * HIDC + l16];
            b.y = sW[(kg + 1) * HIDC + l16];
            c = __builtin_amdgcn_wmma_f32_16x16x4_f32(
                    false, a, false, b, (short)0, c, false, false);
        }
    }

    // D: VGPR r -> row (wave*16 + r + 8*half), col l16
    const int rowt = row0b + wave * 16;
    float* dst = hw + (size_t)rowt * HIDC + (size_t)(8 * half) * HIDC + l16;
    if (rowt + 16 <= N_NODESC) {             // full tile: unguarded stores
#pragma unroll
        for (int r = 0; r < 8; ++r) dst[(size_t)r * HIDC] = c[r];
    } else {
#pragma unroll
        for (int r = 0; r < 8; ++r) {
            int row = rowt + r + 8 * half;
            if (row < N_NODESC) hw[(size_t)row * HIDC + l16] = c[r];
        }
    }
}

// ------------------------------------------------- scatter 1: agg[dst] += norm*hw[src]
__global__ void k_scatter1(const int* __restrict__ esrc, const int* __restrict__ edst,
                           const float* __restrict__ enorm,
                           const float* __restrict__ hw, float* __restrict__ agg) {
    long long gid = (long long)blockIdx.x * blockDim.x + threadIdx.x;
    if (gid >= (long long)N_EDGESC * HIDC) return;
    int e  = (int)(gid >> 4);
    int ch = (int)(gid & 15);
    int s = esrc[e], d = edst[e];
    atomicAdd(&agg[(size_t)d * HIDC + ch], enorm[e] * hw[(size_t)s * HIDC + ch]);
}

// ------------------------------------------- h = agg + dinv^2*hw + b1 ; BN partial sums
__global__ __launch_bounds__(256) void k_bn_pre(const float* __restrict__ dinv,
                                                const float* __restrict__ hw,
                                                const float* __restrict__ b1,
                                                float* __restrict__ agg,  // in: agg, out: h
                                                float* __restrict__ stats) {
    __shared__ float ssum[256], ssq[256];
    long long gid = (long long)blockIdx.x * blockDim.x + threadIdx.x;
    int tid = threadIdx.x;
    float v = 0.0f;
    if (gid < (long long)N_NODESC * HIDC) {
        int i  = (int)(gid >> 4);
        int ch = (int)(gid & 15);
        float di = dinv[i];
        v = agg[gid] + di * di * hw[gid] + b1[ch];
        agg[gid] = v;                               // h stored in place
    }
    ssum[tid] = v;
    ssq[tid]  = v * v;
    __syncthreads();
    // strides are multiples of 16 so the channel (tid & 15) is preserved
    for (int st = 128; st >= 16; st >>= 1) {
        if (tid < st) { ssum[tid] += ssum[tid + st]; ssq[tid] += ssq[tid + st]; }
        __syncthreads();
    }
    if (tid < 16) {
        atomicAdd(&stats[tid],      ssum[tid]);
        atomicAdd(&stats[16 + tid], ssq[tid]);
    }
}

__global__ void k_bn_final(const float* __restrict__ gamma, const float* __restrict__ beta,
                           float* __restrict__ stats) {
    int c = threadIdx.x;
    if (c < HIDC) {
        float mean = stats[c] / (float)N_NODESC;
        float var  = stats[16 + c] / (float)N_NODESC - mean * mean;
        float sc   = gamma[c] * rsqrtf(var + BN_EPSF);
        stats[32 + c] = sc;                  // scale
        stats[48 + c] = beta[c] - mean * sc; // shift
    }
}

// ------------------------------------- BN apply + ReLU + GEMM2 (K=16 -> 2, per node)
__global__ void k_gemm2(const float* __restrict__ h, const float* __restrict__ stats,
                        const float* __restrict__ W2, float* __restrict__ hw2) {
    int i = blockIdx.x * blockDim.x + threadIdx.x;
    if (i >= N_NODESC) return;
    float a0 = 0.0f, a1 = 0.0f;
#pragma unroll
    for (int c = 0; c < HIDC; ++c) {
        float v = h[(size_t)i * HIDC + c] * stats[32 + c] + stats[48 + c];
        v = fmaxf(v, 0.0f);
        a0 += v * W2[c * 2 + 0];
        a1 += v * W2[c * 2 + 1];
    }
    hw2[(size_t)i * 2 + 0] = a0;
    hw2[(size_t)i * 2 + 1] = a1;
}

// ------------------------------------------------- scatter 2 (2 channels)
__global__ void k_scatter2(const int* __restrict__ esrc, const int* __restrict__ edst,
                           const float* __restrict__ enorm,
                           const float* __restrict__ hw2, float* __restrict__ agg2) {
    long long gid = (long long)blockIdx.x * blockDim.x + threadIdx.x;
    if (gid >= (long long)N_EDGESC * NCLSC) return;
    int e  = (int)(gid >> 1);
    int ch = (int)(gid & 1);
    int s = esrc[e], d = edst[e];
    atomicAdd(&agg2[(size_t)d * 2 + ch], enorm[e] * hw2[(size_t)s * 2 + ch]);
}

// ------------------------------------------------- finalize: bias + log_softmax
__global__ void k_out(const float* __restrict__ agg2, const float* __restrict__ dinv,
                      const float* __restrict__ hw2, const float* __restrict__ b2,
                      float* __restrict__ out) {
    int i = blockIdx.x * blockDim.x + threadIdx.x;
    if (i >= N_NODESC) return;
    float di = dinv[i];
    float sl = di * di;
    float z0 = agg2[(size_t)i * 2 + 0] + sl * hw2[(size_t)i * 2 + 0] + b2[0];
    float z1 = agg2[(size_t)i * 2 + 1] + sl * hw2[(size_t)i * 2 + 1] + b2[1];
    float m  = fmaxf(z0, z1);
    float lse = m + logf(expf(z0 - m) + expf(z1 - m));
    out[(size_t)i * 2 + 0] = z0 - lse;
    out[(size_t)i * 2 + 1] = z1 - lse;
}

// ---------------------------------------------------------------- launcher
extern "C" void kernel_launch(void* const* d_in, const int* in_sizes, int n_in,
                              void* d_out, int out_size, void* d_ws, size_t ws_size,
                              hipStream_t stream) {
    const float*     x     = (const float*)d_in[0];
    const long long* ei    = (const long long*)d_in[1];   // int64 per reference
    const float*     W1    = (const float*)d_in[2];
    const float*     b1    = (const float*)d_in[3];
    const float*     gamma = (const float*)d_in[4];
    const float*     beta  = (const float*)d_in[5];
    const float*     W2    = (const float*)d_in[6];
    const float*     b2    = (const float*)d_in[7];
    float* out = (float*)d_out;

    // static workspace layout (256-byte aligned slabs)
    char* ws = (char*)d_ws;
    size_t off = 0;
    float* deg   = (float*)(ws + off); off += ((size_t)N_NODESC * 4 + 255) & ~255ULL;         // deg -> dinv
    float* hw    = (float*)(ws + off); off += ((size_t)N_NODESC * HIDC * 4 + 255) & ~255ULL;  // x@W1
    float* agg   = (float*)(ws + off); off += ((size_t)N_NODESC * HIDC * 4 + 255) & ~255ULL;  // agg -> h
    float* hw2   = (float*)(ws + off); off += ((size_t)N_NODESC * NCLSC * 4 + 255) & ~255ULL; // h@W2
    float* agg2  = (float*)(ws + off); off += ((size_t)N_NODESC * NCLSC * 4 + 255) & ~255ULL;
    float* stats = (float*)(ws + off); off += 256;  // [sum16 | sumsq16 | scale16 | shift16]
    int*   esrc  = (int*)(ws + off);   off += ((size_t)N_EDGESC * 4 + 255) & ~255ULL;
    int*   edst  = (int*)(ws + off);   off += ((size_t)N_EDGESC * 4 + 255) & ~255ULL;
    float* enorm = (float*)(ws + off); off += ((size_t)N_EDGESC * 4 + 255) & ~255ULL;

    const int T = 256;
    // re-initialize every call (graph replay safe)
    k_fill<<<(N_NODESC + T - 1) / T, T, 0, stream>>>(deg, 1.0f, N_NODESC);
    k_fill<<<(N_NODESC * HIDC + T - 1) / T, T, 0, stream>>>(agg, 0.0f, N_NODESC * HIDC);
    k_fill<<<(N_NODESC * NCLSC + T - 1) / T, T, 0, stream>>>(agg2, 0.0f, N_NODESC * NCLSC);
    k_fill<<<1, 64, 0, stream>>>(stats, 0.0f, 64);

    k_degree<<<(N_EDGESC + T - 1) / T, T, 0, stream>>>(ei, deg);
    k_rsqrt<<<(N_NODESC + T - 1) / T, T, 0, stream>>>(deg);   // deg now holds dinv
    k_edgeprep<<<(N_EDGESC + T - 1) / T, T, 0, stream>>>(ei, deg, esrc, edst, enorm);

    // WMMA GEMM1 with async double-buffered A staging
    k_gemm1<<<(N_NODESC + 127) / 128, 256, 0, stream>>>(x, W1, hw);

    {   long long tot = (long long)N_EDGESC * HIDC;
        k_scatter1<<<(unsigned)((tot + T - 1) / T), T, 0, stream>>>(esrc, edst, enorm, hw, agg); }

    {   long long tot = (long long)N_NODESC * HIDC;
        k_bn_pre<<<(unsigned)((tot + T - 1) / T), T, 0, stream>>>(deg, hw, b1, agg, stats); }
    k_bn_final<<<1, 16, 0, stream>>>(gamma, beta, stats);

    k_gemm2<<<(N_NODESC + T - 1) / T, T, 0, stream>>>(agg, stats, W2, hw2);

    {   long long tot = (long long)N_EDGESC * NCLSC;
        k_scatter2<<<(unsigned)((tot + T - 1) / T), T, 0, stream>>>(esrc, edst, enorm, hw2, agg2); }

    k_out<<<(N_NODESC + T - 1) / T, T, 0, stream>>>(agg2, deg, hw2, b2, out);
}